// CuGraphSAGEConv_58342835749307
// MI455X (gfx1250) — compile-verified
//
#include <hip/hip_runtime.h>

typedef __attribute__((ext_vector_type(2))) float v2f;
typedef __attribute__((ext_vector_type(4))) float v4f;
typedef __attribute__((ext_vector_type(8))) float v8f;

#define IN_FEATS   128
#define OUT_FEATS  128
#define KDIM       256          // 2 * IN_FEATS
#define TILE_M     16
#define LDS_STRIDE 260          // 256 + 4 pad; 260 % 64 == 4 -> conflict-free A-frag reads

__global__ __launch_bounds__(256) void sage_fused_wmma_f32(
    const float* __restrict__ feat,     // [n][128]
    const int*   __restrict__ offsets,  // [n+1]
    const int*   __restrict__ indices,  // [E]
    const float* __restrict__ weight,   // [128][256] row-major (out, 2*in)
    const float* __restrict__ bias,     // [128]
    float*       __restrict__ out,      // [n][128]
    int n_nodes, int n_tiles)
{
    __shared__ float Atile[TILE_M][LDS_STRIDE];

    const int tid  = threadIdx.x;
    const int wave = tid >> 5;          // 8 waves
    const int lane = tid & 31;
    const int m    = lane & 15;         // row (A) / col (B,D) within 16
    const int kh   = lane >> 4;         // K-half select per ISA 32-bit layouts
    const int n0   = wave * 16;         // this wave's output column block

    // ---- Preload B = W^T[:, n0:n0+16] into VGPRs (loop-invariant across tiles) ----
    // ISA 32-bit B 4x16 layout: lane l -> N = l%16, VGPR r -> K = (l/16)*2 + r.
    // B(k, n) = W[n][k]  =>  contiguous float2 from W row (n0+m) at col kk*4 + kh*2.
    const float* wrow = weight + (size_t)(n0 + m) * KDIM + (kh << 1);
    v2f Breg[64];
#pragma unroll
    for (int kk = 0; kk < 64; ++kk)
        Breg[kk] = *(const v2f*)(wrow + (kk << 2));

    const float bsc = bias[n0 + m];

    for (int tile = blockIdx.x; tile < n_tiles; tile += (int)gridDim.x) {
        const int node0 = tile * TILE_M;

        // ---- Phase 1: A-tile rows = [self_feat | mean(neighbor feats)] ----
        // Each wave builds 2 rows; each lane covers 4 contiguous floats of D_in.
#pragma unroll
        for (int jj = 0; jj < 2; ++jj) {
            const int j    = wave * 2 + jj;
            const int node = node0 + j;
            v4f selfv = {0.f, 0.f, 0.f, 0.f};
            v4f acc   = {0.f, 0.f, 0.f, 0.f};
            if (node < n_nodes) {
                selfv = *(const v4f*)(feat + (size_t)node * IN_FEATS + lane * 4);
                const int beg = offsets[node];
                const int deg = offsets[node + 1] - beg;
                for (int e = 0; e < deg; ++e) {
                    const int src = indices[beg + e];
                    acc += *(const v4f*)(feat + (size_t)src * IN_FEATS + lane * 4);
                }
                const float inv = 1.0f / (float)(deg > 1 ? deg : 1);
                acc = acc * inv;
            }
            *(v4f*)&Atile[j][lane * 4]            = selfv;  // cols 0..127
            *(v4f*)&Atile[j][IN_FEATS + lane * 4] = acc;    // cols 128..255
        }
        __syncthreads();

        // ---- Phase 2: D(16x16) += A(16x4) x B(4x16) over K = 256, fp32 WMMA ----
        // ISA 32-bit A 16x4 layout: lane l -> M = l%16, VGPR r -> K = (l/16)*2 + r
        // => contiguous float2 from LDS row m at col kk*4 + kh*2.
        v8f c = {0.f, 0.f, 0.f, 0.f, 0.f, 0.f, 0.f, 0.f};
#pragma unroll
        for (int kk = 0; kk < 64; ++kk) {
            const int kcol = (kk << 2) + (kh << 1);
            v2f a = *(const v2f*)&Atile[m][kcol];
            c = __builtin_amdgcn_wmma_f32_16x16x4_f32(
                    /*neg_a=*/false, a, /*neg_b=*/false, Breg[kk],
                    /*c_mod=*/(short)0, c, /*reuse_a=*/false, /*reuse_b=*/false);
        }

        // ---- Store D: lane holds rows M = 8*kh + r (r = VGPR idx), col n0 + m ----
#pragma unroll
        for (int r = 0; r < 8; ++r) {
            const int row = node0 + kh * 8 + r;
            if (row < n_nodes)
                out[(size_t)row * OUT_FEATS + n0 + m] = c[r] + bsc;
        }
        __syncthreads();   // protect Atile before next tile overwrites it
    }
}

extern "C" void kernel_launch(void* const* d_in, const int* in_sizes, int n_in,
                              void* d_out, int out_size, void* d_ws, size_t ws_size,
                              hipStream_t stream) {
    const float* feat    = (const float*)d_in[0];
    const int*   offsets = (const int*)d_in[1];
    const int*   indices = (const int*)d_in[2];
    const float* weight  = (const float*)d_in[3];
    const float* bias    = (const float*)d_in[4];
    float*       out     = (float*)d_out;

    const int n_nodes = in_sizes[0] / IN_FEATS;
    const int n_tiles = (n_nodes + TILE_M - 1) / TILE_M;
    const int grid    = n_tiles < 1024 ? n_tiles : 1024;

    sage_fused_wmma_f32<<<grid, 256, 0, stream>>>(
        feat, offsets, indices, weight, bias, out, n_nodes, n_tiles);
}